// SimpleGatedGCNEdgesLayer_83476984365560
// MI455X (gfx1250) — compile-verified
//
#include <hip/hip_runtime.h>

typedef float v2f __attribute__((ext_vector_type(2)));
typedef float v8f __attribute__((ext_vector_type(8)));

#define NN   50000
#define NE   800000
#define DIM  128
#define EPSV 1e-6f

// ---------------------------------------------------------------------------
// Kernel 1: fused  hn = h*norm ; {Ah,Bh,Dh,Eh} = hn @ W{A,B,D,E} + b
// 1 block = 16 node rows, 8 waves; wave w owns column tile w for all 4 mats.
// Uses V_WMMA_F32_16X16X4_F32 (exact fp32), K-loop of 32 steps.
// ---------------------------------------------------------------------------
__global__ __launch_bounds__(256)
void gemm4_kernel(const float* __restrict__ h, const float* __restrict__ norm,
                  const float* __restrict__ WA, const float* __restrict__ bA,
                  const float* __restrict__ WB, const float* __restrict__ bB,
                  const float* __restrict__ WD, const float* __restrict__ bD,
                  const float* __restrict__ WE, const float* __restrict__ bE,
                  float* __restrict__ Ah, float* __restrict__ Bh,
                  float* __restrict__ Dh, float* __restrict__ Eh)
{
    __shared__ float sA[16 * 132];          // 16x128 hn tile, stride 132 (no bank conflicts)
    const int tid  = threadIdx.x;
    const int row0 = blockIdx.x * 16;

    // stage hn = h * norm into LDS (8 floats / thread)
#pragma unroll
    for (int i = 0; i < 8; ++i) {
        int l = tid + i * 256;
        int r = l >> 7;
        int c = l & 127;
        sA[r * 132 + c] = h[(row0 + r) * DIM + c] * norm[row0 + r];
    }
    __syncthreads();

    const int wid   = tid >> 5;
    const int lane  = tid & 31;
    const int nloc  = lane & 15;
    const int hi    = lane >> 4;            // 0: lanes 0-15, 1: lanes 16-31
    const int col   = wid * 16 + nloc;      // output column
    const int khalf = hi << 1;              // A/B fragment K sub-offset (0 or 2)

    const float* __restrict__ Wp[4] = {WA, WB, WD, WE};
    const float* __restrict__ bp[4] = {bA, bB, bD, bE};
    float*       __restrict__ Op[4] = {Ah, Bh, Dh, Eh};

    v8f acc[4];
#pragma unroll
    for (int p = 0; p < 4; ++p) {
        const float bv = bp[p][col];        // bias broadcast along rows
#pragma unroll
        for (int i = 0; i < 8; ++i) acc[p][i] = bv;
    }

    const float* sArow = &sA[nloc * 132];
#pragma unroll 4
    for (int k0 = 0; k0 < DIM; k0 += 4) {
        // A fragment: 16x4 fp32, lane = 16*(k>=2) + m, vgpr = k&1
        v2f a;
        a.x = sArow[k0 + khalf];
        a.y = sArow[k0 + khalf + 1];
#pragma unroll
        for (int p = 0; p < 4; ++p) {
            // B fragment: 4x16 fp32, vgpr v holds row K = khalf + v, col = lane%16
            v2f b;
            const float* w = Wp[p] + col;
            b.x = w[(k0 + khalf)     * DIM];
            b.y = w[(k0 + khalf + 1) * DIM];
            acc[p] = __builtin_amdgcn_wmma_f32_16x16x4_f32(
                false, a, false, b, (short)0, acc[p], false, false);
        }
    }

    // C/D layout: vgpr i -> row (i + 8*hi), col = lane%16
    const int rbase = row0 + (hi << 3);
#pragma unroll
    for (int p = 0; p < 4; ++p) {
#pragma unroll
        for (int i = 0; i < 8; ++i) {
            Op[p][(rbase + i) * DIM + col] = acc[p][i];
        }
    }
}

// ---------------------------------------------------------------------------
// Kernel 0: zero num/den accumulators (ws is poisoned by harness)
// ---------------------------------------------------------------------------
__global__ __launch_bounds__(256)
void zero_kernel(float4* __restrict__ p, int n4)
{
    int i = blockIdx.x * blockDim.x + threadIdx.x;
    if (i < n4) p[i] = make_float4(0.f, 0.f, 0.f, 0.f);
}

// ---------------------------------------------------------------------------
// Kernel 2: per-edge gate + weighted segment-sum via L2-resident f32 atomics.
// 32 threads per edge, 4 features each (float4 gathers).
// ---------------------------------------------------------------------------
__global__ __launch_bounds__(256)
void edge_kernel(const int* __restrict__ src, const int* __restrict__ dst,
                 const float* __restrict__ Bh, const float* __restrict__ Dh,
                 const float* __restrict__ Eh,
                 float* __restrict__ num, float* __restrict__ den)
{
    int gid = blockIdx.x * blockDim.x + threadIdx.x;
    int e   = gid >> 5;
    if (e >= NE) return;
    int f = (gid & 31) << 2;
    int s = src[e];
    int d = dst[e];

    const float4 dh = *(const float4*)(Dh + s * DIM + f);
    const float4 eh = *(const float4*)(Eh + d * DIM + f);
    const float4 bh = *(const float4*)(Bh + s * DIM + f);

    float s0 = 1.f / (1.f + __expf(-(dh.x + eh.x)));
    float s1 = 1.f / (1.f + __expf(-(dh.y + eh.y)));
    float s2 = 1.f / (1.f + __expf(-(dh.z + eh.z)));
    float s3 = 1.f / (1.f + __expf(-(dh.w + eh.w)));

    float* np = num + d * DIM + f;
    float* dp = den + d * DIM + f;
    unsafeAtomicAdd(np + 0, s0 * bh.x);
    unsafeAtomicAdd(np + 1, s1 * bh.y);
    unsafeAtomicAdd(np + 2, s2 * bh.z);
    unsafeAtomicAdd(np + 3, s3 * bh.w);
    unsafeAtomicAdd(dp + 0, s0);
    unsafeAtomicAdd(dp + 1, s1);
    unsafeAtomicAdd(dp + 2, s2);
    unsafeAtomicAdd(dp + 3, s3);
}

// ---------------------------------------------------------------------------
// Kernel 3: h_out = (Ah + num/(den+eps)) * norm   (Ah lives in d_out, in-place)
// ---------------------------------------------------------------------------
__global__ __launch_bounds__(256)
void finalize_kernel(float* __restrict__ out,
                     const float* __restrict__ num, const float* __restrict__ den,
                     const float* __restrict__ norm)
{
    int i = blockIdx.x * blockDim.x + threadIdx.x;   // one float4 per thread
    if (i >= NN * DIM / 4) return;
    int row = i >> 5;                                 // (i*4)/128
    float nm = norm[row];
    float4 a  = ((const float4*)out)[i];
    float4 nu = ((const float4*)num)[i];
    float4 de = ((const float4*)den)[i];
    float4 r;
    r.x = (a.x + nu.x / (de.x + EPSV)) * nm;
    r.y = (a.y + nu.y / (de.y + EPSV)) * nm;
    r.z = (a.z + nu.z / (de.z + EPSV)) * nm;
    r.w = (a.w + nu.w / (de.w + EPSV)) * nm;
    ((float4*)out)[i] = r;
}

// ---------------------------------------------------------------------------
// Kernel 4: pass-through copy of e into the second output slot
// ---------------------------------------------------------------------------
__global__ __launch_bounds__(256)
void copye_kernel(const float4* __restrict__ e, float4* __restrict__ o, int n4)
{
    int i = blockIdx.x * blockDim.x + threadIdx.x;
    if (i < n4) o[i] = e[i];
}

// ---------------------------------------------------------------------------
extern "C" void kernel_launch(void* const* d_in, const int* in_sizes, int n_in,
                              void* d_out, int out_size, void* d_ws, size_t ws_size,
                              hipStream_t stream)
{
    const float* h    = (const float*)d_in[0];
    const float* e    = (const float*)d_in[1];
    const float* norm = (const float*)d_in[2];
    const int*   src  = (const int*)d_in[3];
    const int*   dst  = (const int*)d_in[4];
    const float* WA   = (const float*)d_in[5];
    const float* bA   = (const float*)d_in[6];
    const float* WB   = (const float*)d_in[7];
    const float* bB   = (const float*)d_in[8];
    const float* WD   = (const float*)d_in[9];
    const float* bD   = (const float*)d_in[10];
    const float* WE   = (const float*)d_in[11];
    const float* bE   = (const float*)d_in[12];

    float* out   = (float*)d_out;            // [0, N*D): h_out ; [N*D, N*D+E*D): e
    float* out_e = out + (size_t)NN * DIM;

    // workspace layout (floats): Bh | Dh | Eh | num | den   (5 * N*D * 4B = 128 MB)
    float* Bh  = (float*)d_ws;
    float* Dh  = Bh + (size_t)NN * DIM;
    float* Eh  = Dh + (size_t)NN * DIM;
    float* num = Eh + (size_t)NN * DIM;
    float* den = num + (size_t)NN * DIM;

    // 0) zero num/den (12500 blocks covers 2*N*D floats as float4)
    zero_kernel<<<12500, 256, 0, stream>>>((float4*)num, 2 * NN * DIM / 4);

    // 1) fused norm + 4 projections (WMMA). Ah goes straight into d_out.
    gemm4_kernel<<<NN / 16, 256, 0, stream>>>(h, norm,
                                              WA, bA, WB, bB, WD, bD, WE, bE,
                                              out, Bh, Dh, Eh);

    // 2) edge gating + atomic segment sums (E*32 threads)
    edge_kernel<<<(NE * 32) / 256, 256, 0, stream>>>(src, dst, Bh, Dh, Eh, num, den);

    // 3) finalize h_out in place
    finalize_kernel<<<(NN * DIM / 4) / 256, 256, 0, stream>>>(out, num, den, norm);

    // 4) e output is identity
    copye_kernel<<<(NE * DIM / 4) / 256, 256, 0, stream>>>((const float4*)e,
                                                           (float4*)out_e,
                                                           NE * DIM / 4);
}